// TopKRankingLoss_73057393705016
// MI455X (gfx1250) — compile-verified
//
#include <hip/hip_runtime.h>
#include <stdint.h>

// TopK ranking loss for MI455X (gfx1250).
// Bandwidth-bound (128 MiB streamed once ~ 5.6us @ 23.3 TB/s). One workgroup
// per row; rows (16 KB each) are DMA'd into LDS by the Tensor Data Mover,
// then an exact 4-pass radix-select runs out of LDS. Bucket selection is
// wave-parallel (wave32 shuffle suffix-scan) to avoid a serial SALU hotspot.

#define COLS   4096
#define BS     256
#define NWAVES (BS / 32)

typedef unsigned int u32;
typedef __attribute__((ext_vector_type(4))) u32 u32x4;
typedef __attribute__((ext_vector_type(8))) u32 u32x8;

// Monotone mapping: larger float  <=>  larger unsigned key.
__device__ __forceinline__ u32 mono_key(float f) {
  u32 u = __float_as_uint(f);
  return (u & 0x80000000u) ? ~u : (u | 0x80000000u);
}

// Issue one TDM descriptor: copy COLS fp32 (one row, 16 KB) global -> LDS.
// D# layout per CDNA5 ISA.
__device__ __forceinline__ void tdm_load_row_to_lds(u32 lds_byte_off,
                                                    const float* gsrc) {
  unsigned long long ga = (unsigned long long)(uintptr_t)gsrc;
  u32x4 g0;
  g0[0] = 1u;                                         // count = 1 (valid D#)
  g0[1] = lds_byte_off;                               // lds_addr (bytes)
  g0[2] = (u32)(ga & 0xFFFFFFFFull);                  // global_addr[31:0]
  g0[3] = ((u32)(ga >> 32) & 0x01FFFFFFu)             // global_addr[56:32]
          | 0x80000000u;                              // type = 2 ("image")
  u32x8 g1;
  g1[0] = (2u << 16);                                 // data_size=2 (4B), mask=0
  g1[1] = ((u32)COLS & 0xFFFFu) << 16;                // tensor_dim0[15:0]
  g1[2] = (((u32)COLS >> 16) & 0xFFFFu) | (1u << 16); // dim0 hi16 | tensor_dim1=1
  g1[3] = ((u32)COLS & 0xFFFFu) << 16;                // tile_dim0 = COLS
  g1[4] = 1u;                                         // tile_dim1=1, tile_dim2=0
  g1[5] = (u32)COLS;                                  // tensor_dim0_stride
  g1[6] = 0u;
  g1[7] = 0u;
  u32x4 gz = {0u, 0u, 0u, 0u};                        // groups 2/3 unused (2-D)
  asm volatile("tensor_load_to_lds %0, %1, %2, %3"
               :: "s"(g0), "s"(g1), "s"(gz), "s"(gz)
               : "memory");
}

__device__ __forceinline__ void wait_tensorcnt0() {
#if __has_builtin(__builtin_amdgcn_s_wait_tensorcnt)
  __builtin_amdgcn_s_wait_tensorcnt(0);
#else
  asm volatile("s_wait_tensorcnt 0x0" ::: "memory");
#endif
}

// Deterministic block reduction (wave32 shuffles + LDS tree).
__device__ __forceinline__ float block_reduce_sum(float v, float* s_red) {
  __syncthreads();  // protect s_red reuse across successive calls
#pragma unroll
  for (int off = 16; off > 0; off >>= 1) v += __shfl_xor(v, off, 32);
  const int lane = threadIdx.x & 31;
  const int wid  = threadIdx.x >> 5;
  if (lane == 0) s_red[wid] = v;
  __syncthreads();
  if (wid == 0) {
    float t = (lane < NWAVES) ? s_red[lane] : 0.0f;
#pragma unroll
    for (int off = 16; off > 0; off >>= 1) t += __shfl_xor(t, off, 32);
    if (lane == 0) s_red[0] = t;
  }
  __syncthreads();
  return s_red[0];
}

// Exact k-th largest (in key space; flip=true => k-th smallest float) among
// valid (non-NaN) targets via 4x8-bit radix select over an LDS histogram.
// Returns threshold key T and r = #elements equal to T that belong to the set.
__device__ void radix_select(const float* s_t, bool flip, u32 k,
                             u32* hist, u32* s_sel, u32& T_out, u32& r_out) {
  u32 prefix = 0, pmask = 0, kk = k;
  for (int pass = 0; pass < 4; ++pass) {
    const int shift = 24 - 8 * pass;
    for (int b = threadIdx.x; b < 256; b += BS) hist[b] = 0;
    __syncthreads();
    for (int i = threadIdx.x; i < COLS; i += BS) {
      float t = s_t[i];
      if (t == t) {  // valid
        u32 key = mono_key(t);
        if (flip) key = ~key;
        if ((key & pmask) == prefix)
          atomicAdd(&hist[(key >> shift) & 0xFFu], 1u);
      }
    }
    __syncthreads();
    // Wave-parallel bucket selection: lane l of wave 0 owns buckets
    // [8l, 8l+8). Suffix-scan of group sums via wave32 shuffles (no
    // barriers), then each lane resolves its 8 buckets locally.
    if (threadIdx.x < 32) {
      const int lane = threadIdx.x;
      u32 c[8];
      u32 gsum = 0;
#pragma unroll
      for (int j = 0; j < 8; ++j) {
        c[j] = hist[lane * 8 + j];
        gsum += c[j];
      }
      // inclusive suffix sum over lanes (sum of gsum for lanes >= lane)
      u32 suf = gsum;
#pragma unroll
      for (int off = 1; off < 32; off <<= 1) {
        u32 o = __shfl_down(suf, off, 32);
        if (lane + off < 32) suf += o;
      }
      u32 cum = suf - gsum;  // count in buckets strictly above this group
#pragma unroll
      for (int j = 7; j >= 0; --j) {
        // bucket b = 8*lane + j; 'cum' = count of elements in buckets > b
        if (cum < kk && cum + c[j] >= kk) {
          s_sel[0] = (u32)(lane * 8 + j);
          s_sel[1] = kk - cum;
        }
        cum += c[j];
      }
    }
    __syncthreads();
    prefix |= s_sel[0] << shift;
    pmask  |= 0xFFu << shift;
    kk = s_sel[1];
  }
  T_out = prefix;
  r_out = kk;
}

// Sum of predictions over { key > T } plus r tie-elements with key == T.
__device__ float selected_sum(const float* s_t, const float* s_p, bool flip,
                              u32 T, u32 r, u32* eq_cnt, float* s_red) {
  if (threadIdx.x == 0) *eq_cnt = 0u;
  __syncthreads();
  float local = 0.0f;
  for (int i = threadIdx.x; i < COLS; i += BS) {
    float t = s_t[i];
    if (t == t) {
      u32 key = mono_key(t);
      if (flip) key = ~key;
      if (key > T) {
        local += s_p[i];
      } else if (key == T) {
        u32 idx = atomicAdd(eq_cnt, 1u);
        if (idx < r) local += s_p[i];
      }
    }
  }
  return block_reduce_sum(local, s_red);
}

__global__ __launch_bounds__(BS) void topk_row_kernel(
    const float* __restrict__ pred, const float* __restrict__ tgt,
    float* __restrict__ wsm, float* __restrict__ wsc) {
  __shared__ float s_tgt[COLS];    // 16 KB
  __shared__ float s_pred[COLS];   // 16 KB
  __shared__ u32   hist[256];
  __shared__ u32   s_sel[2];
  __shared__ u32   s_cnt[2];       // [0]=valid count, [1]=tie counter
  __shared__ float s_red[NWAVES];

  const int row = blockIdx.x;

  // --- Stage both rows into LDS with the Tensor Data Mover (wave 0 issues;
  // TDM is a scalar-descriptor DMA, EXEC-independent, tracked by TENSORcnt).
  if (threadIdx.x < 32) {
    tdm_load_row_to_lds((u32)(uintptr_t)&s_tgt[0],  tgt  + (size_t)row * COLS);
    tdm_load_row_to_lds((u32)(uintptr_t)&s_pred[0], pred + (size_t)row * COLS);
    wait_tensorcnt0();
  }
  if (threadIdx.x == 0) s_cnt[0] = 0u;
  __syncthreads();

  // --- n = count of valid (non-NaN) targets (integer atomics: deterministic)
  u32 local_n = 0;
  for (int i = threadIdx.x; i < COLS; i += BS) {
    float t = s_tgt[i];
    local_n += (t == t) ? 1u : 0u;
  }
#pragma unroll
  for (int off = 16; off > 0; off >>= 1) local_n += __shfl_xor(local_n, off, 32);
  if ((threadIdx.x & 31) == 0) atomicAdd(&s_cnt[0], local_n);
  __syncthreads();
  const int ni = (int)s_cnt[0];

  int k = (int)floorf((float)ni * 0.2f);  // matches fp32 math of the reference
  if (k < 1) k = 1;
  const bool contrib = (ni > 1) && (k < (ni / 2));
  if (!contrib) {
    if (threadIdx.x == 0) { wsm[row] = 0.0f; wsc[row] = 0.0f; }
    return;
  }

  u32 T, r;
  radix_select(s_tgt, false, (u32)k, hist, s_sel, T, r);
  const float top_sum = selected_sum(s_tgt, s_pred, false, T, r, &s_cnt[1], s_red);
  radix_select(s_tgt, true, (u32)k, hist, s_sel, T, r);
  const float bot_sum = selected_sum(s_tgt, s_pred, true, T, r, &s_cnt[1], s_red);

  if (threadIdx.x == 0) {
    const float kf = (float)k;
    float margin = 1.0f - (top_sum / kf - bot_sum / kf);
    wsm[row] = (margin > 0.0f) ? margin : 0.0f;
    wsc[row] = 1.0f;
  }
}

// Deterministic fixed-order final reduction + epilogue.
__global__ __launch_bounds__(BS) void finalize_kernel(
    const float* __restrict__ wsm, const float* __restrict__ wsc,
    float* __restrict__ out, int rows) {
  __shared__ float s_n[NWAVES];
  __shared__ float s_d[NWAVES];
  float num = 0.0f, den = 0.0f;
  for (int i = threadIdx.x; i < rows; i += BS) {
    num += wsm[i];
    den += wsc[i];
  }
#pragma unroll
  for (int off = 16; off > 0; off >>= 1) {
    num += __shfl_xor(num, off, 32);
    den += __shfl_xor(den, off, 32);
  }
  const int lane = threadIdx.x & 31;
  const int wid  = threadIdx.x >> 5;
  if (lane == 0) { s_n[wid] = num; s_d[wid] = den; }
  __syncthreads();
  if (wid == 0) {
    num = (lane < NWAVES) ? s_n[lane] : 0.0f;
    den = (lane < NWAVES) ? s_d[lane] : 0.0f;
#pragma unroll
    for (int off = 16; off > 0; off >>= 1) {
      num += __shfl_xor(num, off, 32);
      den += __shfl_xor(den, off, 32);
    }
    if (lane == 0) out[0] = (den > 0.0f) ? (num / fmaxf(den, 1.0f)) : 0.0f;
  }
}

extern "C" void kernel_launch(void* const* d_in, const int* in_sizes, int n_in,
                              void* d_out, int out_size, void* d_ws, size_t ws_size,
                              hipStream_t stream) {
  const float* pred = (const float*)d_in[0];
  const float* tgt  = (const float*)d_in[1];
  float* out = (float*)d_out;
  const int rows = in_sizes[0] / COLS;  // 4096

  float* wsm = (float*)d_ws;        // per-row margin * contrib
  float* wsc = wsm + rows;          // per-row contrib flag

  topk_row_kernel<<<rows, BS, 0, stream>>>(pred, tgt, wsm, wsc);
  finalize_kernel<<<1, BS, 0, stream>>>(wsm, wsc, out, rows);
}